// KGDAL_11888469475553
// MI455X (gfx1250) — compile-verified
//
#include <hip/hip_runtime.h>

// Problem constants
#define B_ 512
#define T_ 512
#define L_ 8
#define G_ 32          // 4*L gate width

typedef __attribute__((ext_vector_type(2))) float v2f;
typedef __attribute__((ext_vector_type(8))) float v8f;

// ---------------------------------------------------------------------------
// Kernel 1: xz = x @ W + b   ([B*T, F] x [F, 32] -> [B*T, 32])
// f32 WMMA 16x16x4. One wave per 16-row M tile, two N tiles (0..15, 16..31).
// A layout (ISA 7.12.2, 32-bit A 16x4): lanes 0-15 -> K=k0,k0+1 ; lanes 16-31
// -> K=k0+2,k0+3, M = lane&15. B rows striped across lanes per VGPR.
// ---------------------------------------------------------------------------
template<int F>
__global__ __launch_bounds__(256)
void xz_gemm(const float* __restrict__ x, const float* __restrict__ W,
             const float* __restrict__ bias, float* __restrict__ xz)
{
    const int lane = threadIdx.x & 31;
    const int wave = threadIdx.x >> 5;
    const int l15  = lane & 15;
    const int half = lane >> 4;                 // K-pair selector
    const int m0   = (blockIdx.x * 8 + wave) * 16;

    v8f c0 = {};
    v8f c1 = {};
    #pragma unroll
    for (int k0 = 0; k0 < F; k0 += 4) {
        const int ka = k0 + 2 * half;
        v2f a;
        a.x = x[(size_t)(m0 + l15) * F + ka];
        a.y = x[(size_t)(m0 + l15) * F + ka + 1];
        v2f b0;                                  // B tile cols 0..15
        b0.x = W[(size_t)ka * G_ + l15];
        b0.y = W[(size_t)(ka + 1) * G_ + l15];
        v2f b1;                                  // B tile cols 16..31
        b1.x = W[(size_t)ka * G_ + 16 + l15];
        b1.y = W[(size_t)(ka + 1) * G_ + 16 + l15];
        c0 = __builtin_amdgcn_wmma_f32_16x16x4_f32(false, a, false, b0,
                                                   (short)0, c0, false, false);
        c1 = __builtin_amdgcn_wmma_f32_16x16x4_f32(false, a, false, b1,
                                                   (short)0, c1, false, false);
    }
    const float bias0 = bias[l15];
    const float bias1 = bias[16 + l15];
    #pragma unroll
    for (int r = 0; r < 8; ++r) {               // C/D: row = r + 8*half, col = l15
        const int row = m0 + r + 8 * half;
        xz[(size_t)row * G_ + l15]      = c0[r] + bias0;
        xz[(size_t)row * G_ + 16 + l15] = c1[r] + bias1;
    }
}

// ---------------------------------------------------------------------------
// Kernel 2: LSTM recurrence. One wave (32 lanes = 32 gates) handles 4 batch
// rows; h (8 wide) lives in lanes 0..7 and is broadcast with shuffles.
// blockIdx.y selects the branch slot; gate order i,f,g,o (g uses relu).
// Writes h into merged[b,t, 8*branch + j] (merged aliases d_out comb region).
// ---------------------------------------------------------------------------
__global__ __launch_bounds__(32)
void lstm_rec(const float* __restrict__ xz0, const float* __restrict__ xz1,
              const float* __restrict__ xz2, const float* __restrict__ xz3,
              const float* __restrict__ U0,  const float* __restrict__ U1,
              const float* __restrict__ U2,  const float* __restrict__ U3,
              float* __restrict__ merged, int branchBase)
{
    const int slot = blockIdx.y;
    const float* xz = (slot == 0) ? xz0 : (slot == 1) ? xz1 : (slot == 2) ? xz2 : xz3;
    const float* U  = (slot == 0) ? U0  : (slot == 1) ? U1  : (slot == 2) ? U2  : U3;
    const int off = (branchBase + slot) * L_;
    const int g  = threadIdx.x;
    const int b0 = blockIdx.x * 4;

    float u[8];
    #pragma unroll
    for (int j = 0; j < 8; ++j) u[j] = U[j * G_ + g];
    const bool isRelu = (g >= 16) && (g < 24);   // the "g" gate
    const int jj = g & 7;

    float h[4] = {0.f, 0.f, 0.f, 0.f};
    float c[4] = {0.f, 0.f, 0.f, 0.f};
    for (int t = 0; t < T_; ++t) {
        #pragma unroll
        for (int r = 0; r < 4; ++r) {
            const size_t base = ((size_t)(b0 + r) * T_ + t) * G_;
            float z = xz[base + g];
            #pragma unroll
            for (int j = 0; j < 8; ++j)
                z += __shfl(h[r], j, 32) * u[j];
            const float s = 1.f / (1.f + __expf(-z));
            const float a = isRelu ? fmaxf(z, 0.f) : s;
            const float iv = __shfl(a, jj,      32);
            const float fv = __shfl(a, jj + 8,  32);
            const float gv = __shfl(a, jj + 16, 32);
            const float ov = __shfl(a, jj + 24, 32);
            const float cn = fv * c[r] + iv * gv;
            const float hn = ov * fmaxf(cn, 0.f);
            c[r] = cn;
            h[r] = hn;
            if (g < 8) merged[base + off + g] = hn;
        }
    }
}

// ---------------------------------------------------------------------------
// Kernel 3: fused attention + head, one block per batch row.
// s_f = tanh(sum_t m*w_f); scale_j = exp(sf_j)*SC_j / (exp(sf_0)*SC_0)
// (softmax denominator cancels in the ratio). s_t = tanh(m@w_t); a_t =
// softmax_T. comb = a_t*scale written IN PLACE over merged (d_out region),
// sent = sum_t m*comb, y = sigmoid(sent@w_out).
// ---------------------------------------------------------------------------
__global__ __launch_bounds__(256)
void attention(float* __restrict__ out,
               const float* __restrict__ w_f, const float* __restrict__ b_f,
               const float* __restrict__ w_t, const float* __restrict__ b_t,
               const float* __restrict__ w_o, const float* __restrict__ b_o)
{
    __shared__ float wf[T_];
    __shared__ float wt[G_];
    __shared__ float part[8][G_];
    __shared__ float st[T_];
    __shared__ float red[256];
    __shared__ float scl[G_];

    const int tid = threadIdx.x;
    const int b   = blockIdx.x;
    float* mrow = out + B_ + (size_t)b * T_ * G_;

    wf[tid] = w_f[tid];
    wf[tid + 256] = w_f[tid + 256];
    if (tid < G_) wt[tid] = w_t[tid];
    __syncthreads();

    const int p = tid >> 5;     // time partition 0..7
    const int j = tid & 31;     // feature 0..31

    // ---- feature attention scores ----
    float acc = 0.f;
    for (int t = p; t < T_; t += 8)
        acc += mrow[(size_t)t * G_ + j] * wf[t];
    part[p][j] = acc;
    __syncthreads();
    if (tid < G_) {
        float s = 0.f;
        #pragma unroll
        for (int q = 0; q < 8; ++q) s += part[q][tid];
        const float sf = tanhf(s + b_f[0]);
        const float SC[4] = {0.0215f, 0.33634f, 0.68064f, 0.97682f};
        const float ef = __expf(sf) * SC[tid >> 3];
        const float e0 = __shfl(ef, 0, 32);
        scl[tid] = ef / e0;
    }
    __syncthreads();

    // ---- time attention scores (2 timesteps per thread) ----
    float stv[2];
    #pragma unroll
    for (int q = 0; q < 2; ++q) {
        const int t = tid + q * 256;
        const float4* r4 = (const float4*)(mrow + (size_t)t * G_);
        float d = 0.f;
        #pragma unroll
        for (int k = 0; k < 8; ++k) {
            const float4 v = r4[k];
            d += v.x * wt[4 * k] + v.y * wt[4 * k + 1] +
                 v.z * wt[4 * k + 2] + v.w * wt[4 * k + 3];
        }
        stv[q] = tanhf(d + b_t[0]);
        st[t] = stv[q];
    }
    red[tid] = fmaxf(stv[0], stv[1]);
    __syncthreads();
    for (int s = 128; s > 0; s >>= 1) {
        if (tid < s) red[tid] = fmaxf(red[tid], red[tid + s]);
        __syncthreads();
    }
    const float mx = red[0];
    __syncthreads();
    const float e0 = __expf(stv[0] - mx);
    const float e1 = __expf(stv[1] - mx);
    st[tid] = e0;
    st[tid + 256] = e1;
    red[tid] = e0 + e1;
    __syncthreads();
    for (int s = 128; s > 0; s >>= 1) {
        if (tid < s) red[tid] += red[tid + s];
        __syncthreads();
    }
    const float inv = 1.f / red[0];
    __syncthreads();

    // ---- comb (in place) + sent + head ----
    const float sj = scl[j];
    float sacc = 0.f;
    for (int t = p; t < T_; t += 8) {
        const size_t idx = (size_t)t * G_ + j;
        const float m  = mrow[idx];
        const float cb = st[t] * inv * sj;
        mrow[idx] = cb;                       // attention_comb output
        sacc += m * cb;
    }
    part[p][j] = sacc;
    __syncthreads();
    if (tid < G_) {
        float s = 0.f;
        #pragma unroll
        for (int q = 0; q < 8; ++q) s += part[q][tid];
        float r = s * w_o[tid];
        #pragma unroll
        for (int o = 16; o > 0; o >>= 1) r += __shfl_down(r, o, 32);
        if (tid == 0) out[b] = 1.f / (1.f + __expf(-(r + b_o[0])));
    }
}

// ---------------------------------------------------------------------------
extern "C" void kernel_launch(void* const* d_in, const int* in_sizes, int n_in,
                              void* d_out, int out_size, void* d_ws, size_t ws_size,
                              hipStream_t stream)
{
    const float* x[4]  = {(const float*)d_in[0],  (const float*)d_in[1],
                          (const float*)d_in[2],  (const float*)d_in[3]};
    const float* W[4]  = {(const float*)d_in[4],  (const float*)d_in[7],
                          (const float*)d_in[10], (const float*)d_in[13]};
    const float* U[4]  = {(const float*)d_in[5],  (const float*)d_in[8],
                          (const float*)d_in[11], (const float*)d_in[14]};
    const float* bb[4] = {(const float*)d_in[6],  (const float*)d_in[9],
                          (const float*)d_in[12], (const float*)d_in[15]};
    const float* w_f = (const float*)d_in[16];
    const float* b_f = (const float*)d_in[17];
    const float* w_t = (const float*)d_in[18];
    const float* b_t = (const float*)d_in[19];
    const float* w_o = (const float*)d_in[20];
    const float* b_o = (const float*)d_in[21];

    float* out    = (float*)d_out;
    float* merged = out + B_;                  // alias attention_comb region
    float* ws     = (float*)d_ws;
    const size_t XZ = (size_t)B_ * T_ * G_;    // xz elements per branch
    const int gemmBlocks = (B_ * T_ / 16) / 8; // 2048 blocks * 8 waves

    const bool parallel = ws_size >= 4 * XZ * sizeof(float);
    if (parallel) {
        xz_gemm<64><<<gemmBlocks, 256, 0, stream>>>(x[0], W[0], bb[0], ws + 0 * XZ);
        xz_gemm<32><<<gemmBlocks, 256, 0, stream>>>(x[1], W[1], bb[1], ws + 1 * XZ);
        xz_gemm<48><<<gemmBlocks, 256, 0, stream>>>(x[2], W[2], bb[2], ws + 2 * XZ);
        xz_gemm<24><<<gemmBlocks, 256, 0, stream>>>(x[3], W[3], bb[3], ws + 3 * XZ);
        lstm_rec<<<dim3(B_ / 4, 4), 32, 0, stream>>>(
            ws, ws + XZ, ws + 2 * XZ, ws + 3 * XZ,
            U[0], U[1], U[2], U[3], merged, 0);
    } else {
        for (int br = 0; br < 4; ++br) {
            switch (br) {
                case 0: xz_gemm<64><<<gemmBlocks, 256, 0, stream>>>(x[0], W[0], bb[0], ws); break;
                case 1: xz_gemm<32><<<gemmBlocks, 256, 0, stream>>>(x[1], W[1], bb[1], ws); break;
                case 2: xz_gemm<48><<<gemmBlocks, 256, 0, stream>>>(x[2], W[2], bb[2], ws); break;
                default: xz_gemm<24><<<gemmBlocks, 256, 0, stream>>>(x[3], W[3], bb[3], ws); break;
            }
            lstm_rec<<<dim3(B_ / 4, 1), 32, 0, stream>>>(
                ws, ws, ws, ws, U[br], U[br], U[br], U[br], merged, br);
        }
    }
    attention<<<B_, 256, 0, stream>>>(out, w_f, b_f, w_t, b_t, w_o, b_o);
}